// BoxListComposeHNMS_49658411876614
// MI455X (gfx1250) — compile-verified
//
#include <hip/hip_runtime.h>
#include <hip/hip_bf16.h>
#include <stdint.h>

// ---------------- constants matching the reference ----------------
#define HN_W0     16.0f
#define HN_H0     16.0f
#define HN_AW0    9.6f          // ALPHA * W0 = 0.6f*16.0f (same f32 value as JAX)
#define HN_AH0    9.6f
#define HN_GAMMA  1.4f
#define HN_IOU    0.5f
#define HN_NEG    (-1e30f)

#define CAP_LOG2  21
#define CAP       (1u << CAP_LOG2)          // 2,097,152 slots (load factor < 0.5)
#define EMPTY_KEY 0xFFFFFFFFFFFFFFFFULL     // keyB quantized coords >= 0 => never all-ones

// ---------------- helpers ----------------
__device__ __forceinline__ uint32_t ordf(float f) {
    uint32_t b = __float_as_uint(f);
    return (b & 0x80000000u) ? ~b : (b | 0x80000000u);
}
__device__ __forceinline__ float unordf(uint32_t u) {
    uint32_t b = (u & 0x80000000u) ? (u & 0x7FFFFFFFu) : ~u;
    return __uint_as_float(b);
}
__device__ __forceinline__ unsigned long long mix64(unsigned long long x) {
    x ^= x >> 33; x *= 0xff51afd7ed558ccdULL;
    x ^= x >> 33; x *= 0xc4ceb9fe1a85ec53ULL;
    x ^= x >> 33;
    return x;
}

__device__ __forceinline__ unsigned long long box_key(float4 r, float dw, float dh,
                                                      float dx, float dy) {
    const float invlg = 1.0f / logf(HN_GAMMA);
    float qw = floorf(logf(r.z / HN_W0) * invlg + dw);
    float qh = floorf(logf(r.w / HN_H0) * invlg + dh);
    float qx = floorf(r.x / (HN_AW0 * powf(HN_GAMMA, qw)) + dx);
    float qy = floorf(r.y / (HN_AH0 * powf(HN_GAMMA, qh)) + dy);
    int iqw = (int)qw, iqh = (int)qh, iqx = (int)qx, iqy = (int)qy;
    uint32_t keyA = ((uint32_t)(iqw & 0xFFFF) << 16) | (uint32_t)(iqh & 0xFFFF);
    uint32_t keyB = ((uint32_t)(iqx & 0xFFFF) << 16) | (uint32_t)(iqy & 0xFFFF);
    return ((unsigned long long)keyA << 32) | (unsigned long long)keyB;
}

__device__ __forceinline__ float iou_cxywh(float4 a, float4 b) {
    float ax1 = a.x - 0.5f * a.z, ay1 = a.y - 0.5f * a.w;
    float ax2 = a.x + 0.5f * a.z, ay2 = a.y + 0.5f * a.w;
    float bx1 = b.x - 0.5f * b.z, by1 = b.y - 0.5f * b.w;
    float bx2 = b.x + 0.5f * b.z, by2 = b.y + 0.5f * b.w;
    float iw = fmaxf(fminf(ax2, bx2) - fmaxf(ax1, bx1), 0.0f);
    float ih = fmaxf(fminf(ay2, by2) - fmaxf(ay1, by1), 0.0f);
    float inter = iw * ih;
    float uni = a.z * a.w + b.z * b.w - inter;
    return inter / fmaxf(uni, 1e-12f);
}

// ---------------- kernels ----------------
// Clear two 16B table entries per thread (B128 stores).
__global__ void hn_clear_table(ulonglong2* __restrict__ tkeys,
                               ulonglong2* __restrict__ tvals, int cap2) {
    int i = blockIdx.x * blockDim.x + threadIdx.x;
    if (i < cap2) {
        ulonglong2 e; e.x = EMPTY_KEY; e.y = EMPTY_KEY;
        ulonglong2 z; z.x = 0ULL;      z.y = 0ULL;
        tkeys[i] = e;
        tvals[i] = z;
    }
}

// Insert each box into the hash table and atomicMax its packed (score,idx).
__global__ void hn_insert(const float4* __restrict__ rects,
                          const float* __restrict__ scores,
                          const unsigned* __restrict__ keep1, int useMs,
                          const float* __restrict__ offRow,
                          unsigned long long* __restrict__ tkeys,
                          unsigned long long* __restrict__ tvals,
                          unsigned* __restrict__ slots, int n) {
    int i = blockIdx.x * blockDim.x + threadIdx.x;
    if (i >= n) return;
    // gfx1250 speculative prefetch of the linear input streams (global_prefetch_b8)
    __builtin_prefetch(&rects[i + 4096], 0, 0);
    __builtin_prefetch(&scores[i + 16384], 0, 0);
    float dw = offRow[0], dh = offRow[1], dx = offRow[2], dy = offRow[3];
    float4 r = rects[i];
    unsigned long long key = box_key(r, dw, dh, dx, dy);
    float v = scores[i];
    if (useMs && !keep1[i]) v = HN_NEG;
    unsigned long long packed =
        ((unsigned long long)ordf(v) << 32) | (unsigned long long)(~(unsigned)i);

    unsigned h = (unsigned)mix64(key) & (CAP - 1u);
    for (;;) {
        unsigned long long cur = tkeys[h];
        if (cur == key) break;
        if (cur == EMPTY_KEY) {
            unsigned long long prev = atomicCAS(&tkeys[h], EMPTY_KEY, key);
            if (prev == EMPTY_KEY || prev == key) break;
        }
        h = (h + 1u) & (CAP - 1u);
    }
    atomicMax(&tvals[h], packed);
    slots[i] = h;
}

// keep1 round: box kept iff it is its bucket's argmax.
// Packed values are unique, so comparing the embedded index (~i in low 32b) suffices.
__global__ void hn_check_argmax(const unsigned* __restrict__ slots,
                                const unsigned long long* __restrict__ tvals,
                                unsigned* __restrict__ keepA, int first, int n) {
    int i = blockIdx.x * blockDim.x + threadIdx.x;
    if (i >= n) return;
    unsigned winner = (unsigned)(tvals[slots[i]] & 0xFFFFFFFFull);
    unsigned res = (winner == ~(unsigned)i) ? 1u : 0u;
    keepA[i] = first ? res : (keepA[i] & res);
}

// rerank round: kept iff rep, or IoU with bucket representative <= thr.
// On the last round this also emits the final packed (score,idx) array,
// fusing what used to be a separate build_packed pass.
__global__ void hn_rerank_check(const float4* __restrict__ rects,
                                const unsigned* __restrict__ slots,
                                const unsigned long long* __restrict__ tvals,
                                unsigned* __restrict__ keepB,
                                const unsigned* __restrict__ keepA,
                                const float* __restrict__ scores,
                                unsigned long long* __restrict__ packed,
                                int first, int last, int n) {
    int i = blockIdx.x * blockDim.x + threadIdx.x;
    if (i >= n) return;
    unsigned long long rep = tvals[slots[i]];
    unsigned repIdx = ~(unsigned)(rep & 0xFFFFFFFFull);
    unsigned res;
    if (repIdx == (unsigned)i) {
        res = 1u;
    } else {
        float4 a = rects[i];
        float4 b = rects[repIdx];
        res = (iou_cxywh(a, b) <= HN_IOU) ? 1u : 0u;
    }
    unsigned kb = first ? res : (keepB[i] & res);
    if (last) {
        float v = (keepA[i] & kb) ? scores[i] : HN_NEG;
        packed[i] = ((unsigned long long)ordf(v) << 32)
                  | (unsigned long long)(~(unsigned)i);
    } else {
        keepB[i] = kb;
    }
}

// ---------------- 64-bit radix-select (top-K) ----------------
struct SelState {
    unsigned long long prefix;
    unsigned k;
    unsigned count;
    unsigned hist[256];
};

__global__ void hn_sel_init(SelState* st, unsigned k) {
    int t = threadIdx.x;
    if (t == 0) { st->prefix = 0ULL; st->k = k; st->count = 0u; }
    st->hist[t] = 0u;
}

__global__ void hn_sel_pass(const unsigned long long* __restrict__ packed, int n,
                            SelState* st, int pass) {
    __shared__ unsigned lh[256];
    int t = threadIdx.x;
    for (int b = t; b < 256; b += blockDim.x) lh[b] = 0u;
    __syncthreads();
    unsigned long long prefix = st->prefix;
    int shift = 56 - 8 * pass;
    int stride = gridDim.x * blockDim.x;
    for (int i = blockIdx.x * blockDim.x + t; i < n; i += stride) {
        // gfx1250 speculative prefetch of the stream ahead (global_prefetch_b8)
        __builtin_prefetch(&packed[i + 8192], 0, 0);
        unsigned long long e = packed[i];
        bool match = (pass == 0) || ((e >> (shift + 8)) == prefix);
        if (match) atomicAdd(&lh[(unsigned)(e >> shift) & 255u], 1u);
    }
    __syncthreads();
    for (int b = t; b < 256; b += blockDim.x)
        if (lh[b]) atomicAdd(&st->hist[b], lh[b]);
}

__global__ void hn_sel_scan(SelState* st) {
    __shared__ unsigned h[256];
    int t = threadIdx.x;
    h[t] = st->hist[t];
    __syncthreads();
    if (t == 0) {
        unsigned k = st->k;
        unsigned long long prefix = st->prefix;
        for (int b = 255; b >= 0; --b) {
            unsigned c = h[b];
            if (k <= c) { prefix = (prefix << 8) | (unsigned long long)(unsigned)b; break; }
            k -= c;
        }
        st->k = k;
        st->prefix = prefix;
    }
    __syncthreads();
    st->hist[t] = 0u;   // ready for next pass
}

__global__ void hn_collect(const unsigned long long* __restrict__ packed, int n,
                           SelState* st, unsigned long long* __restrict__ coll, int cap) {
    int i = blockIdx.x * blockDim.x + threadIdx.x;
    if (i >= n) return;
    unsigned long long e = packed[i];
    if (e >= st->prefix) {               // exactly K elements (packed keys are unique)
        unsigned p = atomicAdd(&st->count, 1u);
        if ((int)p < cap) coll[p] = e;
    }
}

// One block of 1024 threads: bitonic sort (descending) in LDS, then emit rows.
// Output is never re-read: use non-temporal stores (gfx1250 TH=NT cache hint).
__global__ void hn_sort_emit(const unsigned long long* __restrict__ coll,
                             const float4* __restrict__ rects,
                             float* __restrict__ out, int K) {
    __shared__ unsigned long long s[1024];
    int t = threadIdx.x;
    s[t] = (t < K) ? coll[t] : 0ULL;     // 0 sorts to the tail (all real keys have ~idx != 0)
    __syncthreads();
    for (int k = 2; k <= 1024; k <<= 1) {
        for (int j = k >> 1; j > 0; j >>= 1) {
            int ixj = t ^ j;
            if (ixj > t) {
                unsigned long long a = s[t], b = s[ixj];
                bool desc = ((t & k) == 0);
                if (desc ? (a < b) : (a > b)) { s[t] = b; s[ixj] = a; }
            }
            __syncthreads();
        }
    }
    if (t < K) {
        unsigned long long e = s[t];
        unsigned idx = ~(unsigned)(e & 0xFFFFFFFFull);
        float val = unordf((unsigned)(e >> 32));
        float4 r = rects[idx];
        __builtin_nontemporal_store(r.x, &out[t * 5 + 0]);
        __builtin_nontemporal_store(r.y, &out[t * 5 + 1]);
        __builtin_nontemporal_store(r.z, &out[t * 5 + 2]);
        __builtin_nontemporal_store(r.w, &out[t * 5 + 3]);
        __builtin_nontemporal_store(val, &out[t * 5 + 4]);
    }
}

// ---------------- launcher ----------------
extern "C" void kernel_launch(void* const* d_in, const int* in_sizes, int n_in,
                              void* d_out, int out_size, void* d_ws, size_t ws_size,
                              hipStream_t stream) {
    const float4* rects  = (const float4*)d_in[0];
    const float*  scores = (const float*)d_in[1];
    const float*  off1   = (const float*)d_in[2];
    const float*  off2   = (const float*)d_in[3];
    const int N    = in_sizes[0] / 4;
    const int R1   = in_sizes[2] / 4;       // 4 offset rows
    const int R2   = in_sizes[3] / 4;       // 4 offset rows
    const int K    = out_size / 5;          // 1000
    float* out = (float*)d_out;

    // workspace layout (256B aligned sections)
    uintptr_t base = (uintptr_t)d_ws;
    auto align256 = [](uintptr_t p) { return (p + 255) & ~(uintptr_t)255; };
    uintptr_t p = align256(base);
    unsigned long long* tkeys = (unsigned long long*)p; p = align256(p + (size_t)CAP * 8);
    unsigned long long* tvals = (unsigned long long*)p; p = align256(p + (size_t)CAP * 8);
    unsigned* slots           = (unsigned*)p;           p = align256(p + (size_t)N * 4);
    unsigned* keepA           = (unsigned*)p;           p = align256(p + (size_t)N * 4);
    unsigned* keepB           = (unsigned*)p;           p = align256(p + (size_t)N * 4);
    unsigned long long* packed= (unsigned long long*)p; p = align256(p + (size_t)N * 8);
    SelState* st              = (SelState*)p;           p = align256(p + sizeof(SelState));
    unsigned long long* coll  = (unsigned long long*)p; p = align256(p + 1024 * 8);
    (void)ws_size;

    const int BLK = 256;
    const int gN  = (N + BLK - 1) / BLK;
    const int gC2 = ((int)(CAP / 2) + BLK - 1) / BLK;

    // ---- phase 1: keep1 = AND over offsets1 rows of per-bucket argmax ----
    for (int r = 0; r < R1; ++r) {
        hn_clear_table<<<gC2, BLK, 0, stream>>>((ulonglong2*)tkeys, (ulonglong2*)tvals,
                                                (int)(CAP / 2));
        hn_insert<<<gN, BLK, 0, stream>>>(rects, scores, keepA, /*useMs=*/0,
                                          off1 + r * 4, tkeys, tvals, slots, N);
        hn_check_argmax<<<gN, BLK, 0, stream>>>(slots, tvals, keepA,
                                                /*first=*/(r == 0), N);
    }

    // ---- phase 2: keep2 = AND over offsets2 rows of rerank-vs-representative ----
    // Final round fuses the packed-score build (keep = keepA & keepB).
    for (int r = 0; r < R2; ++r) {
        hn_clear_table<<<gC2, BLK, 0, stream>>>((ulonglong2*)tkeys, (ulonglong2*)tvals,
                                                (int)(CAP / 2));
        hn_insert<<<gN, BLK, 0, stream>>>(rects, scores, keepA, /*useMs=*/1,
                                          off2 + r * 4, tkeys, tvals, slots, N);
        hn_rerank_check<<<gN, BLK, 0, stream>>>(rects, slots, tvals, keepB,
                                                keepA, scores, packed,
                                                /*first=*/(r == 0),
                                                /*last=*/(r == R2 - 1), N);
    }

    // ---- phase 3: 64-bit radix top-K over packed, then sort + emit ----
    hn_sel_init<<<1, 256, 0, stream>>>(st, (unsigned)K);
    for (int pass = 0; pass < 8; ++pass) {
        hn_sel_pass<<<1024, BLK, 0, stream>>>(packed, N, st, pass);
        hn_sel_scan<<<1, 256, 0, stream>>>(st);
    }
    hn_collect<<<gN, BLK, 0, stream>>>(packed, N, st, coll, 1024);
    hn_sort_emit<<<1, 1024, 0, stream>>>(coll, rects, out, K);
}